// StandardGCN_43980465111673
// MI455X (gfx1250) — compile-verified
//
#include <hip/hip_runtime.h>

typedef __attribute__((ext_vector_type(2))) float v2f;
typedef __attribute__((ext_vector_type(8))) float v8f;

// ---------------------------------------------------------------------------
// Degree / normalization
// ---------------------------------------------------------------------------
__global__ void k_deg_ones(float* __restrict__ deg, int n) {
    int i = blockIdx.x * blockDim.x + threadIdx.x;
    if (i < n) deg[i] = 1.0f;                      // self-loop contributes 1
}

__global__ void k_deg_edges(const int* __restrict__ dst, float* __restrict__ deg, int E) {
    int e = blockIdx.x * blockDim.x + threadIdx.x;
    if (e < E) atomicAdd(&deg[dst[e]], 1.0f);
}

__global__ void k_make_dis(const float* __restrict__ deg, float* __restrict__ dis, int n) {
    int i = blockIdx.x * blockDim.x + threadIdx.x;
    if (i < n) dis[i] = rsqrtf(deg[i]);
}

// ---------------------------------------------------------------------------
// Dense GEMM  C[n,128] = X[n,128] @ W[128,128]  via V_WMMA_F32_16X16X4_F32.
// One block = 16 output rows (staged in LDS) x 128 cols; 8 waves, each wave
// owns a 16x16 tile and iterates K in steps of 4 (32 WMMA ops).
// Fragment layout (ISA 7.12.2, wave32):
//   A 16x4:  lanes 0-15 rows M=0..15 {K, K+1}; lanes 16-31 {K+2, K+3}
//   B 4x16:  lanes 0-15 cols N=0..15 {K, K+1}; lanes 16-31 {K+2, K+3}
//   C 16x16: VGPR v -> M = v (lanes 0-15) / v+8 (lanes 16-31), N = lane&15
// ---------------------------------------------------------------------------
__global__ __launch_bounds__(256) void k_gemm128_wmma(
    const float* __restrict__ X, const float* __restrict__ W,
    float* __restrict__ C)
{
    __shared__ float Alds[16 * 128];
    const int rowBase = blockIdx.x * 16;
    const int tid = threadIdx.x;
    for (int i = tid; i < 16 * 128; i += 256) {
        const int r = i >> 7, c = i & 127;
        Alds[i] = X[(size_t)(rowBase + r) * 128 + c];
    }
    __syncthreads();

    const int wave  = tid >> 5;
    const int lane  = tid & 31;
    const int lrow  = lane & 15;          // A row (M) / B,C col (N) within tile
    const int col   = wave * 16 + lrow;   // global column
    const int khalf = (lane >> 4) << 1;   // lanes 16-31 carry K+2,K+3

    v8f acc = {};
#pragma unroll 4
    for (int k = 0; k < 128; k += 4) {
        const int kk = k + khalf;
        v2f a, b;
        a.x = Alds[lrow * 128 + kk];
        a.y = Alds[lrow * 128 + kk + 1];
        b.x = W[(size_t)kk * 128 + col];
        b.y = W[(size_t)(kk + 1) * 128 + col];
        acc = __builtin_amdgcn_wmma_f32_16x16x4_f32(
                  /*neg_a=*/false, a, /*neg_b=*/false, b,
                  /*c_mod=*/(short)0, acc, /*reuse_a=*/false, /*reuse_b=*/false);
    }

    const int rofs = (lane >> 4) << 3;    // lanes 16-31 hold M = v + 8
#pragma unroll
    for (int v = 0; v < 8; ++v)
        C[(size_t)(rowBase + rofs + v) * 128 + col] = acc[v];
}

// ---------------------------------------------------------------------------
// Aggregation: AGG = b + dis^2 * H  (bias + self-loop), then edge atomics.
// ---------------------------------------------------------------------------
__global__ void k_agg_init128(const float* __restrict__ H, const float* __restrict__ b,
                              const float* __restrict__ dis, float* __restrict__ AGG,
                              int n)
{
    long long idx = (long long)blockIdx.x * blockDim.x + threadIdx.x;
    if (idx >= (long long)n * 128) return;
    int i = (int)(idx >> 7), j = (int)(idx & 127);
    float d = dis[i];
    AGG[idx] = b[j] + d * d * H[idx];
}

__global__ void k_agg_edges128(const float* __restrict__ H, float* __restrict__ AGG,
                               const int* __restrict__ src, const int* __restrict__ dst,
                               const float* __restrict__ dis, int E)
{
    long long idx = (long long)blockIdx.x * blockDim.x + threadIdx.x;
    int e = (int)(idx >> 5);              // 32 threads per edge
    if (e >= E) return;
    int j = (int)(idx & 31) << 2;         // 4 floats per thread
    int s = src[e], d = dst[e];
    float w = dis[s] * dis[d];
    const float4 hv = *(const float4*)(H + (size_t)s * 128 + j);
    float* base = AGG + (size_t)d * 128 + j;
    atomicAdd(base + 0, hv.x * w);
    atomicAdd(base + 1, hv.y * w);
    atomicAdd(base + 2, hv.z * w);
    atomicAdd(base + 3, hv.w * w);
}

__global__ void k_relu(float* __restrict__ A, long long n) {
    long long idx = (long long)blockIdx.x * blockDim.x + threadIdx.x;
    if (idx < n) A[idx] = fmaxf(A[idx], 0.0f);
}

// ---------------------------------------------------------------------------
// Layer 3: OUT = 2.  T[i,:2] = H[i,:] @ W3; then init + edge atomics on d_out.
// ---------------------------------------------------------------------------
__global__ void k_gemm_out2(const float* __restrict__ H, const float* __restrict__ W3,
                            float* __restrict__ T, int n)
{
    int i = blockIdx.x * blockDim.x + threadIdx.x;
    if (i >= n) return;
    float a0 = 0.f, a1 = 0.f;
    const float* h = H + (size_t)i * 128;
#pragma unroll 8
    for (int k = 0; k < 128; ++k) {
        float hv = h[k];
        a0 += hv * W3[k * 2 + 0];
        a1 += hv * W3[k * 2 + 1];
    }
    T[i * 2 + 0] = a0;
    T[i * 2 + 1] = a1;
}

__global__ void k_out_init2(const float* __restrict__ T, const float* __restrict__ b3,
                            const float* __restrict__ dis, float* __restrict__ OUT, int n)
{
    int idx = blockIdx.x * blockDim.x + threadIdx.x;
    if (idx >= n * 2) return;
    int i = idx >> 1, o = idx & 1;
    float d = dis[i];
    OUT[idx] = b3[o] + d * d * T[idx];
}

__global__ void k_out_edges2(const float* __restrict__ T, float* __restrict__ OUT,
                             const int* __restrict__ src, const int* __restrict__ dst,
                             const float* __restrict__ dis, int E)
{
    int e = blockIdx.x * blockDim.x + threadIdx.x;
    if (e >= E) return;
    int s = src[e], d = dst[e];
    float w = dis[s] * dis[d];
    atomicAdd(&OUT[d * 2 + 0], T[s * 2 + 0] * w);
    atomicAdd(&OUT[d * 2 + 1], T[s * 2 + 1] * w);
}

// ---------------------------------------------------------------------------
extern "C" void kernel_launch(void* const* d_in, const int* in_sizes, int n_in,
                              void* d_out, int out_size, void* d_ws, size_t ws_size,
                              hipStream_t stream)
{
    const float* x   = (const float*)d_in[0];
    const int*   ei  = (const int*)d_in[1];
    const float* W1  = (const float*)d_in[2];
    const float* b1  = (const float*)d_in[3];
    const float* W2  = (const float*)d_in[4];
    const float* b2  = (const float*)d_in[5];
    const float* W3  = (const float*)d_in[6];
    const float* b3  = (const float*)d_in[7];
    float* out = (float*)d_out;

    const int N = in_sizes[0] / 128;     // 50000
    const int E = in_sizes[1] / 2;       // 800000
    const int* src = ei;                 // edge_index[0]
    const int* dst = ei + E;             // edge_index[1]

    // Workspace layout (floats)
    float* ws   = (float*)d_ws;
    float* deg  = ws;                         // N
    float* dis  = ws + N;                     // N
    float* bufA = ws + 2 * (size_t)N;         // N*128 (GEMM out)
    float* bufB = bufA + (size_t)N * 128;     // N*128 (agg / next input)
    float* t3   = bufB + (size_t)N * 128;     // N*2

    const int TPB = 256;
    const long long nf128 = (long long)N * 128;
    const int gNodes  = (N + TPB - 1) / TPB;
    const int gEdges  = (E + TPB - 1) / TPB;
    const int gElems  = (int)((nf128 + TPB - 1) / TPB);
    const int gEdgeF4 = (int)(((long long)E * 32 + TPB - 1) / TPB);
    const int gTiles  = N / 16;               // 3125, N divisible by 16

    // --- normalization -----------------------------------------------------
    k_deg_ones <<<gNodes, TPB, 0, stream>>>(deg, N);
    k_deg_edges<<<gEdges, TPB, 0, stream>>>(dst, deg, E);
    k_make_dis <<<gNodes, TPB, 0, stream>>>(deg, dis, N);

    // --- layer 1 -----------------------------------------------------------
    k_gemm128_wmma<<<gTiles, TPB, 0, stream>>>(x, W1, bufA);
    k_agg_init128 <<<gElems, TPB, 0, stream>>>(bufA, b1, dis, bufB, N);
    k_agg_edges128<<<gEdgeF4, TPB, 0, stream>>>(bufA, bufB, src, dst, dis, E);
    k_relu        <<<gElems, TPB, 0, stream>>>(bufB, nf128);

    // --- layer 2 -----------------------------------------------------------
    k_gemm128_wmma<<<gTiles, TPB, 0, stream>>>(bufB, W2, bufA);
    k_agg_init128 <<<gElems, TPB, 0, stream>>>(bufA, b2, dis, bufB, N);
    k_agg_edges128<<<gEdgeF4, TPB, 0, stream>>>(bufA, bufB, src, dst, dis, E);
    k_relu        <<<gElems, TPB, 0, stream>>>(bufB, nf128);

    // --- layer 3 (OUT = 2) -------------------------------------------------
    k_gemm_out2<<<gNodes, TPB, 0, stream>>>(bufB, W3, t3, N);
    k_out_init2<<<(2 * N + TPB - 1) / TPB, TPB, 0, stream>>>(t3, b3, dis, out, N);
    k_out_edges2<<<gEdges, TPB, 0, stream>>>(t3, out, src, dst, dis, E);
}